// ChamfersDistance_76974403879473
// MI455X (gfx1250) — compile-verified
//
#include <hip/hip_runtime.h>
#include <math.h>

// CDNA5 (gfx1250) wave32 WMMA types
typedef float v2f __attribute__((ext_vector_type(2)));
typedef float v8f __attribute__((ext_vector_type(8)));

#define BATCH 32
#define NPTS  2048

// One wave32 owns TWO 16-row tiles (32 rows) of "query" points in one batch
// and sweeps all M "reference" points 16 columns at a time.
//
// Per column tile, with V_WMMA_F32_16X16X4_F32 (D = A x B + C, C = 0):
//   A row  = (a0, a1, a2, 1)            (K=0..3)
//   B col  = (-2b0, -2b1, -2b2, |b|^2)
// so D[n,m] = |b_m|^2 - 2 a_n.b_m. The row-constant |a_n|^2 is added once
// AFTER the min-reduction (min over columns commutes with a per-row shift).
//
// ISA 32-bit A 16x4 layout: lanes 0-15 hold K=0 (VGPR0) / K=1 (VGPR1);
// lanes 16-31 hold K=2 / K=3. B mirrors over K with col = lane&15.
// C/D layout: VGPR j -> row j (lanes 0-15) / row 8+j (lanes 16-31), col = lane&15.
__global__ __launch_bounds__(256) void chamfer_rowmin_wmma(
    const float* __restrict__ A_pts,   // [B, N, 3] rows
    const float* __restrict__ B_pts,   // [B, M, 3] cols
    float* __restrict__ rowmin_out,    // [B, N] min_m d(a_n, b_m)
    int N, int M)
{
    const int lane = threadIdx.x & 31;
    const int wave = (blockIdx.x * blockDim.x + threadIdx.x) >> 5;
    const int tilesPerBatch = N >> 5;                 // 32 rows per wave
    const int batch   = wave / tilesPerBatch;
    const int rowbase = (wave % tilesPerBatch) << 5;

    const int  r  = lane & 15;
    const bool hi = lane >= 16;

    // ---- load both A tiles (once per wave)
    const float* abase = A_pts + (size_t)batch * N * 3;
    const float* ap0 = abase + (size_t)(rowbase + r) * 3;
    const float* ap1 = abase + (size_t)(rowbase + 16 + r) * 3;
    const float a00 = ap0[0], a01 = ap0[1], a02 = ap0[2];
    const float a10 = ap1[0], a11 = ap1[1], a12 = ap1[2];
    const float an0 = a00 * a00 + a01 * a01 + a02 * a02;
    const float an1 = a10 * a10 + a11 * a11 + a12 * a12;
    v2f Av0, Av1;
    Av0.x = hi ? a02 : a00;  Av0.y = hi ? 1.0f : a01;   // (K2:K0, K3:K1)
    Av1.x = hi ? a12 : a10;  Av1.y = hi ? 1.0f : a11;

    // |a|^2 indexed by OUTPUT row of the D tile (one-time ds_bpermute)
    const int rowoff = hi ? 8 : 0;
    float anrow0[8], anrow1[8];
#pragma unroll
    for (int j = 0; j < 8; ++j) {
        anrow0[j] = __shfl(an0, rowoff + j, 16);
        anrow1[j] = __shfl(an1, rowoff + j, 16);
    }

    float rmin0[8], rmin1[8];
#pragma unroll
    for (int j = 0; j < 8; ++j) { rmin0[j] = 3.402823466e+38f; rmin1[j] = 3.402823466e+38f; }

    const v8f Cz = {};   // loop-invariant zero accumulator

    const float* bbase = B_pts + (size_t)batch * M * 3;
#pragma unroll 4
    for (int cm = 0; cm < M; cm += 16) {
        const float* bp = bbase + (size_t)(cm + r) * 3;
        __builtin_prefetch(bp + 16 * 3, 0, 1);          // speculative: next tile
        const float b0 = bp[0], b1 = bp[1], b2 = bp[2];
        const float bn = b0 * b0 + b1 * b1 + b2 * b2;
        v2f Bv;
        Bv.x = hi ? (-2.0f * b2) : (-2.0f * b0);        // K=2 : K=0
        Bv.y = hi ?           bn : (-2.0f * b1);        // K=3 : K=1

        v8f D0 = __builtin_amdgcn_wmma_f32_16x16x4_f32(
            false, Av0, false, Bv, (short)0, Cz, false, false);
        v8f D1 = __builtin_amdgcn_wmma_f32_16x16x4_f32(
            false, Av1, false, Bv, (short)0, Cz, false, false);

#pragma unroll
        for (int j = 0; j < 8; ++j) {
            rmin0[j] = fminf(rmin0[j], D0[j]);
            rmin1[j] = fminf(rmin1[j], D1[j]);
        }
    }

    // butterfly min across the 16 column slots within each half-wave
#pragma unroll
    for (int off = 1; off < 16; off <<= 1) {
#pragma unroll
        for (int j = 0; j < 8; ++j) {
            rmin0[j] = fminf(rmin0[j], __shfl_xor(rmin0[j], off, 16));
            rmin1[j] = fminf(rmin1[j], __shfl_xor(rmin1[j], off, 16));
        }
    }

    // lane 0 -> rows 0..7, lane 16 -> rows 8..15 of each tile; add |a|^2, clamp
    if (r == 0) {
        float* o0 = rowmin_out + (size_t)batch * N + rowbase + rowoff;
        float* o1 = o0 + 16;
#pragma unroll
        for (int j = 0; j < 8; ++j) {
            o0[j] = fmaxf(rmin0[j] + anrow0[j], 0.0f);
            o1[j] = fmaxf(rmin1[j] + anrow1[j], 0.0f);
        }
    }
}

// Per-batch: max( mean(sqrt(dist0)), mean(sqrt(dist1)) )
__global__ __launch_bounds__(256) void chamfer_batch_loss(
    const float* __restrict__ dist0, const float* __restrict__ dist1,
    float* __restrict__ loss, int N, int M)
{
    __shared__ float sh[256];
    const int b = blockIdx.x, tid = threadIdx.x;
    float s0 = 0.0f, s1 = 0.0f;
    for (int i = tid; i < N; i += 256) s0 += sqrtf(dist0[(size_t)b * N + i]);
    for (int i = tid; i < M; i += 256) s1 += sqrtf(dist1[(size_t)b * M + i]);

    sh[tid] = s0; __syncthreads();
    for (int s = 128; s > 0; s >>= 1) { if (tid < s) sh[tid] += sh[tid + s]; __syncthreads(); }
    const float m0 = sh[0] / (float)N;
    __syncthreads();

    sh[tid] = s1; __syncthreads();
    for (int s = 128; s > 0; s >>= 1) { if (tid < s) sh[tid] += sh[tid + s]; __syncthreads(); }
    if (tid == 0) loss[b] = fmaxf(m0, sh[0] / (float)M);
}

// mean over batches -> scalar
__global__ void chamfer_mean(const float* __restrict__ loss, float* __restrict__ out, int B)
{
    float v = (threadIdx.x < B) ? loss[threadIdx.x] : 0.0f;
#pragma unroll
    for (int off = 16; off > 0; off >>= 1) v += __shfl_xor(v, off, 32);
    if (threadIdx.x == 0) out[0] = v / (float)B;
}

extern "C" void kernel_launch(void* const* d_in, const int* in_sizes, int n_in,
                              void* d_out, int out_size, void* d_ws, size_t ws_size,
                              hipStream_t stream) {
    (void)in_sizes; (void)n_in; (void)out_size; (void)ws_size;
    const float* input1 = (const float*)d_in[0];  // [B, N, 3]
    const float* input2 = (const float*)d_in[1];  // [B, M, 3]
    const int B = BATCH, N = NPTS, M = NPTS;

    // workspace: dist0 [B*N] | dist1 [B*M] | loss [B]   (~512 KB + 128 B)
    float* dist0 = (float*)d_ws;
    float* dist1 = dist0 + (size_t)B * N;
    float* loss  = dist1 + (size_t)B * M;

    const int waves  = B * (N / 32);        // 2048 wave32s per pass (32 rows each)
    const int blocks = waves * 32 / 256;    // 256 blocks of 8 waves

    chamfer_rowmin_wmma<<<blocks, 256, 0, stream>>>(input1, input2, dist0, N, M);
    chamfer_rowmin_wmma<<<blocks, 256, 0, stream>>>(input2, input1, dist1, M, N);
    chamfer_batch_loss<<<B, 256, 0, stream>>>(dist0, dist1, loss, N, M);
    chamfer_mean<<<1, 32, 0, stream>>>(loss, (float*)d_out, B);
}